// GNN_37177236914713
// MI455X (gfx1250) — compile-verified
//
#include <hip/hip_runtime.h>
#include <hip/hip_bf16.h>

typedef __attribute__((ext_vector_type(2))) float v2f;
typedef __attribute__((ext_vector_type(8))) float v8f;

#define GNN_N 100000
#define GNN_E 1600000
#define GNN_D 64

// ---------------------------------------------------------------- zero buffer
__global__ __launch_bounds__(256) void gnn_zero_kernel(float* __restrict__ p, int n4) {
  int i = blockIdx.x * 256 + threadIdx.x;
  if (i < n4) ((float4*)p)[i] = make_float4(0.f, 0.f, 0.f, 0.f);
}

// ----------------------------------------------------- per-dst in-degree count
__global__ __launch_bounds__(256) void gnn_count_kernel(const int* __restrict__ ei,
                                                        float* __restrict__ cnt) {
  int e = blockIdx.x * 256 + threadIdx.x;
  if (e < GNN_E) {
    int d = ei[GNN_E + e];
    unsafeAtomicAdd(&cnt[d], 1.0f);
  }
}

// -------------------------------------------------------------- inv = 1/max(c,1)
__global__ __launch_bounds__(256) void gnn_inv_kernel(const float* __restrict__ cnt,
                                                      float* __restrict__ inv) {
  int i = blockIdx.x * 256 + threadIdx.x;
  if (i < GNN_N) inv[i] = 1.0f / fmaxf(cnt[i], 1.0f);
}

// ------------------------------------------- scatter-add: agg[dst] += feat[src]
// 16 threads per edge, float4 per thread (4 f32 atomics). Edge index + feature
// rows are L2-resident; atomics resolve in L2.
__global__ __launch_bounds__(256) void gnn_scatter_kernel(const int* __restrict__ ei,
                                                          const float* __restrict__ feat,
                                                          float* __restrict__ agg) {
  int tid = blockIdx.x * 256 + threadIdx.x;
  if (tid >= GNN_E * 16) return;
  int e = tid >> 4;
  int q = (tid & 15) << 2;       // column offset 0..60 step 4
  int s = ei[e];
  int d = ei[GNN_E + e];
  float4 v = *(const float4*)(feat + (size_t)s * GNN_D + q);
  float* o = agg + (size_t)d * GNN_D + q;
  unsafeAtomicAdd(o + 0, v.x);
  unsafeAtomicAdd(o + 1, v.y);
  unsafeAtomicAdd(o + 2, v.z);
  unsafeAtomicAdd(o + 3, v.w);
}

// ---------------------------------------------------------------- fused GEMM
// Y[n,:] = act( (agg[n,:]*inv[n]) @ Wl^T + b + X[n,:] @ Wr^T )
// One wave32 per 16-row tile, full 64 output columns (4 col tiles of 16).
// Uses V_WMMA_F32_16X16X4_F32: A frag v2f (row=lane&15, k=4*kt+(lane>>4)*2),
// B frag v2f (col=lane&15, same k pair, B[k][j]=W[j][k]), C/D v8f.
__global__ __launch_bounds__(256) void gnn_sage_gemm_kernel(
    const float* __restrict__ A,    // aggregated features [N,64]
    const float* __restrict__ X,    // self features [N,64]
    const float* __restrict__ inv,  // [N] 1/max(deg,1)
    const float* __restrict__ Wl,   // [64,64] row-major
    const float* __restrict__ Wr,   // [64,64] row-major
    const float* __restrict__ bias, // [64]
    float* __restrict__ Y,          // [N,64]
    int nRowTiles, int doRelu) {
  int wave = (blockIdx.x * 256 + threadIdx.x) >> 5;
  if (wave >= nRowTiles) return;          // wave-uniform: EXEC stays all-ones
  int lane = threadIdx.x & 31;
  int m    = lane & 15;                   // A row / B col within tile
  int half = lane >> 4;                   // k sub-pair select
  int row0 = wave * 16;
  int row  = row0 + m;
  float scale = inv[row];

  v8f c0 = {}, c1 = {}, c2 = {}, c3 = {};

#pragma unroll 1
  for (int pass = 0; pass < 2; ++pass) {
    const float* M = pass ? X : A;
    const float* W = pass ? Wr : Wl;
    float s = pass ? 1.0f : scale;
    const float* arow = M + (size_t)row * GNN_D;
#pragma unroll
    for (int kt = 0; kt < 16; ++kt) {
      int k0 = kt * 4 + half * 2;
      v2f a = *(const v2f*)(arow + k0);
      a.x *= s; a.y *= s;
      v2f bA = *(const v2f*)(W + (size_t)(0 * 16 + m) * GNN_D + k0);
      v2f bB = *(const v2f*)(W + (size_t)(1 * 16 + m) * GNN_D + k0);
      v2f bC = *(const v2f*)(W + (size_t)(2 * 16 + m) * GNN_D + k0);
      v2f bD = *(const v2f*)(W + (size_t)(3 * 16 + m) * GNN_D + k0);
      c0 = __builtin_amdgcn_wmma_f32_16x16x4_f32(false, a, false, bA, (short)0, c0, false, false);
      c1 = __builtin_amdgcn_wmma_f32_16x16x4_f32(false, a, false, bB, (short)0, c1, false, false);
      c2 = __builtin_amdgcn_wmma_f32_16x16x4_f32(false, a, false, bC, (short)0, c2, false, false);
      c3 = __builtin_amdgcn_wmma_f32_16x16x4_f32(false, a, false, bD, (short)0, c3, false, false);
    }
  }

  // epilogue: bias (+ReLU) and store. C layout: VGPR r -> row r + half*8, col m.
  v8f acc[4] = {c0, c1, c2, c3};
#pragma unroll
  for (int jt = 0; jt < 4; ++jt) {
    int j = jt * 16 + m;
    float bv = bias[j];
#pragma unroll
    for (int r = 0; r < 8; ++r) {
      float v = acc[jt][r] + bv;
      if (doRelu) v = fmaxf(v, 0.0f);
      Y[(size_t)(row0 + r + half * 8) * GNN_D + j] = v;
    }
  }
}

// ------------------------------------------------------------------- launcher
extern "C" void kernel_launch(void* const* d_in, const int* in_sizes, int n_in,
                              void* d_out, int out_size, void* d_ws, size_t ws_size,
                              hipStream_t stream) {
  const float* x   = (const float*)d_in[0];
  const int*   ei  = (const int*)d_in[1];   // [2,E] (src row 0, dst row 1)
  const float* W1l = (const float*)d_in[2];
  const float* b1  = (const float*)d_in[3];
  const float* W1r = (const float*)d_in[4];
  const float* W2l = (const float*)d_in[5];
  const float* b2  = (const float*)d_in[6];
  const float* W2r = (const float*)d_in[7];
  float* out = (float*)d_out;

  // workspace layout
  float* agg = (float*)d_ws;                       // N*64 floats
  float* h   = agg + (size_t)GNN_N * GNN_D;        // N*64 floats
  float* cnt = h + (size_t)GNN_N * GNN_D;          // N floats
  float* inv = cnt + GNN_N;                        // N floats

  const int feat4    = (GNN_N * GNN_D) / 4;        // float4 count
  const int zeroGrid = (feat4 + 255) / 256;
  const int cnt4     = GNN_N / 4;
  const int cntGrid  = (cnt4 + 255) / 256;
  const int edgeGrid = (GNN_E + 255) / 256;
  const int scatGrid = (GNN_E * 16 + 255) / 256;
  const int nGrid    = (GNN_N + 255) / 256;
  const int rowTiles = GNN_N / 16;                 // 6250
  const int gemmGrid = (rowTiles + 7) / 8;         // 8 waves/block

  // degree counts (shared by both layers)
  gnn_zero_kernel<<<cntGrid, 256, 0, stream>>>(cnt, cnt4);
  gnn_count_kernel<<<edgeGrid, 256, 0, stream>>>(ei, cnt);
  gnn_inv_kernel<<<nGrid, 256, 0, stream>>>(cnt, inv);

  // layer 1
  gnn_zero_kernel<<<zeroGrid, 256, 0, stream>>>(agg, feat4);
  gnn_scatter_kernel<<<scatGrid, 256, 0, stream>>>(ei, x, agg);
  gnn_sage_gemm_kernel<<<gemmGrid, 256, 0, stream>>>(agg, x, inv, W1l, W1r, b1, h,
                                                     rowTiles, /*relu=*/1);

  // layer 2
  gnn_zero_kernel<<<zeroGrid, 256, 0, stream>>>(agg, feat4);
  gnn_scatter_kernel<<<scatGrid, 256, 0, stream>>>(ei, h, agg);
  gnn_sage_gemm_kernel<<<gemmGrid, 256, 0, stream>>>(agg, h, inv, W2l, W2r, b2, out,
                                                     rowTiles, /*relu=*/0);
}